// TrackingLoss_20753281974668
// MI455X (gfx1250) — compile-verified
//
#include <hip/hip_runtime.h>
#include <hip/hip_bf16.h>

// ---------------------------------------------------------------------------
// TrackingLoss on MI455X (gfx1250, wave32).
// Pipeline:
//   1) prep_n    : project src/target gaussians, conic opacity, pack WMMA B rows
//   2) prep_m    : pack WMMA A rows for particles
//   3) knn_part  : WMMA f32 16x16x4 distance tiles + streaming per-row top-8,
//                  split-N grid (16 slices) for occupancy -> partial lists
//   4) knn_merge : merge 16 partial sorted lists per particle, radius mask
//   5) score     : depth-sort 8 neighbors, alpha blending, per-particle error
//   6) reduce    : mean over M
// ---------------------------------------------------------------------------

#define W_IMG 960.0f
#define H_IMG 540.0f
#define FXY   500.0f
#define KNN   8
#define RADIUS2 16.0f          // R=4 -> R^2
#define SEG_LIFT 8.0f          // 2*R
#define WAVES_PB 8             // waves per block in knn kernel
#define SPLIT    16            // N-dimension grid split for occupancy

typedef float v2f __attribute__((ext_vector_type(2)));
typedef float v8f __attribute__((ext_vector_type(8)));

__device__ __forceinline__ void project_point(const float* __restrict__ P,
                                              float x, float y, float z,
                                              float& px, float& py,
                                              float& depth, float& reg) {
  // unh_j = x*P[0][j] + y*P[1][j] + z*P[2][j] + P[3][j]  (row-major 4x4)
  float u0 = fmaf(x, P[0], fmaf(y, P[4], fmaf(z, P[8],  P[12])));
  float u1 = fmaf(x, P[1], fmaf(y, P[5], fmaf(z, P[9],  P[13])));
  float u2 = fmaf(x, P[2], fmaf(y, P[6], fmaf(z, P[10], P[14])));
  reg      = fmaf(x, P[3], fmaf(y, P[7], fmaf(z, P[11], P[15])));
  float inv = 1.0f / (reg + 1e-4f);
  px = ((u0 * inv + 1.0f) * W_IMG - 1.0f) * 0.5f;
  py = ((u1 * inv + 1.0f) * H_IMG - 1.0f) * 0.5f;
  depth = u2;
}

// ---- stage 1: per-gaussian preprocessing ----------------------------------
__global__ __launch_bounds__(256) void tl_prep_n(
    const float* __restrict__ srcP, const float* __restrict__ tgtP,
    const float* __restrict__ srcX, const float* __restrict__ tgtX,
    const float* __restrict__ opac, const float* __restrict__ scal,
    const int*   __restrict__ seg,
    float* __restrict__ bvec,   // [N][4] = (-2px,-2py,-2pz, |p|^2)
    float* __restrict__ m2s,    // [N][2]
    float* __restrict__ dsrc,   // [N]
    float* __restrict__ co3,    // [N][3] = (cx/det, cy/det, opacity)
    float* __restrict__ m2t,    // [N][2]
    float* __restrict__ dtgt,   // [N]
    int N) {
  int i = blockIdx.x * blockDim.x + threadIdx.x;
  if (i >= N) return;
  float x = srcX[3*i], y = srcX[3*i+1], z = srcX[3*i+2];
  float px, py, dep, reg;
  project_point(srcP, x, y, z, px, py, dep, reg);
  m2s[2*i] = px; m2s[2*i+1] = py; dsrc[i] = dep;

  float s0  = scal[3*i];
  float cov = s0 * s0;
  float invr = 1.0f / (reg + 1e-4f);
  float fr = FXY * invr;
  float cx = fmaf(cov, fr*fr, 0.3f);
  float cy = fmaf(cov, fr*fr, 0.3f);     // FX==FY -> same factor
  float det = cx * cy;
  co3[3*i]   = cx / det;
  co3[3*i+1] = cy / det;
  co3[3*i+2] = opac[i];

  float pz = SEG_LIFT * (float)seg[i];
  bvec[4*i]   = -2.0f * px;
  bvec[4*i+1] = -2.0f * py;
  bvec[4*i+2] = -2.0f * pz;
  bvec[4*i+3] = fmaf(px, px, fmaf(py, py, pz*pz));

  float tx = tgtX[3*i], ty = tgtX[3*i+1], tz = tgtX[3*i+2];
  float qx, qy, qd, qr;
  project_point(tgtP, tx, ty, tz, qx, qy, qd, qr);
  m2t[2*i] = qx; m2t[2*i+1] = qy; dtgt[i] = qd;
}

// ---- stage 2: per-particle A rows -----------------------------------------
__global__ __launch_bounds__(256) void tl_prep_m(
    const float* __restrict__ sp, const int* __restrict__ pseg,
    float* __restrict__ avec,   // [M][4] = (x, y, 2R*seg, 1)
    int M) {
  int m = blockIdx.x * blockDim.x + threadIdx.x;
  if (m >= M) return;
  avec[4*m]   = sp[2*m];
  avec[4*m+1] = sp[2*m+1];
  avec[4*m+2] = SEG_LIFT * (float)pseg[m];
  avec[4*m+3] = 1.0f;
}

// ---- stage 3: WMMA distance tiles + streaming top-8 (partial) --------------
// Grid: (M/16, SPLIT). Block = 256 threads (8 waves), owns 16 particles and
// the candidate slice [blockIdx.y*N/SPLIT, (blockIdx.y+1)*N/SPLIT); wave w
// streams 1/8 of that slice. Requires N % (SPLIT*WAVES_PB*16) == 0
// (131072 % 2048 == 0) and M % 16 == 0.
__global__ __launch_bounds__(256) void tl_knn_part(
    const float* __restrict__ avec, const float* __restrict__ bvec,
    float* __restrict__ partD,   // [M/16][SPLIT][16][KNN]
    int*   __restrict__ partI,
    int N) {
  __shared__ __align__(32) float tileLDS[WAVES_PB][16*16];   // 8 KB
  __shared__ float mergeD[16][WAVES_PB*2][KNN];              // 8 KB
  __shared__ int   mergeI[16][WAVES_PB*2][KNN];              // 8 KB

  const int tid  = threadIdx.x;
  const int wave = tid >> 5;
  const int lane = tid & 31;
  const int r16  = lane & 15;   // particle row (A) == candidate col (B/C)
  const int half = lane >> 4;   // K-half for A/B, row-half for C
  const int pbase = blockIdx.x * 16;

  // A tile (16x4 f32): lanes 0-15 hold K=0,1 ; lanes 16-31 hold K=2,3
  v2f a;
  {
    const float* ap = avec + (size_t)(pbase + r16) * 4 + half * 2;
    a.x = ap[0]; a.y = ap[1];
  }

  const int sliceN = N / SPLIT;                 // candidates per block
  const int waveN  = sliceN / WAVES_PB;         // candidates per wave
  const int cstart = blockIdx.y * sliceN + wave * waveN;
  const int cend   = cstart + waveN;

  float bd[KNN]; int bi[KNN];
#pragma unroll
  for (int k = 0; k < KNN; ++k) { bd[k] = 3.0e38f; bi[k] = 0; }

  float* myTile = &tileLDS[wave][0];

  // software pipeline: preload first B tile
  v2f bn;
  {
    const float* bp = bvec + (size_t)(cstart + r16) * 4 + half * 2;
    bn.x = bp[0]; bn.y = bp[1];
  }

  for (int c0 = cstart; c0 < cend; c0 += 16) {
    v2f b = bn;
    // issue next tile's load before consuming this one (clamped, branch-free)
    int cn = (c0 + 16 < cend) ? (c0 + 16) : c0;
    const float* bpn = bvec + (size_t)(cn + r16) * 4 + half * 2;
    bn.x = bpn[0]; bn.y = bpn[1];
    __builtin_prefetch(bpn + 128, 0, 1);   // +2 tiles -> global_prefetch_b8

    v8f c = {};
    // D = A x B : d2' = |p2|^2 - 2*p1.p2   (per-row |p1|^2 added at merge)
    c = __builtin_amdgcn_wmma_f32_16x16x4_f32(
        /*neg_a=*/false, a, /*neg_b=*/false, b,
        /*c_mod=*/(short)0, c, /*reuse_a=*/false, /*reuse_b=*/false);

    // Spill C column-major: lane owns col=r16, rows half*8..half*8+7 -> one
    // contiguous 32B chunk (2x ds_store_b128).
    *reinterpret_cast<v8f*>(myTile + r16 * 16 + half * 8) = c;
    __builtin_amdgcn_wave_barrier();
    asm volatile("s_wait_dscnt 0x0" ::: "memory");

    // Selection: lane owns particle-row r16, scans cols half*8..half*8+7.
#pragma unroll
    for (int j = 0; j < 8; ++j) {
      int col = half * 8 + j;
      float v = myTile[col * 16 + r16];
      int idxc = c0 + col;
      if (v < bd[KNN-1]) {
        bd[KNN-1] = v; bi[KNN-1] = idxc;
#pragma unroll
        for (int t = KNN-1; t > 0; --t) {
          if (bd[t] < bd[t-1]) {
            float fsw = bd[t]; bd[t] = bd[t-1]; bd[t-1] = fsw;
            int isw = bi[t]; bi[t] = bi[t-1]; bi[t-1] = isw;
          }
        }
      }
    }
    __builtin_amdgcn_wave_barrier();
  }

  // Publish per-lane lists, merge 16 -> 1 per row, store partial result.
#pragma unroll
  for (int k = 0; k < KNN; ++k) {
    mergeD[r16][wave*2 + half][k] = bd[k];
    mergeI[r16][wave*2 + half][k] = bi[k];
  }
  __syncthreads();

  if (tid < 16) {
    float fd[KNN]; int fi[KNN];
#pragma unroll
    for (int k = 0; k < KNN; ++k) { fd[k] = 3.0e38f; fi[k] = 0; }
    for (int s = 0; s < WAVES_PB*2; ++s) {
      for (int k = 0; k < KNN; ++k) {
        float v = mergeD[tid][s][k];
        if (v >= fd[KNN-1]) break;       // partial lists are sorted ascending
        int idxc = mergeI[tid][s][k];
        fd[KNN-1] = v; fi[KNN-1] = idxc;
#pragma unroll
        for (int t = KNN-1; t > 0; --t) {
          if (fd[t] < fd[t-1]) {
            float fsw = fd[t]; fd[t] = fd[t-1]; fd[t-1] = fsw;
            int isw = fi[t]; fi[t] = fi[t-1]; fi[t-1] = isw;
          }
        }
      }
    }
    size_t pofs = (((size_t)blockIdx.x * SPLIT + blockIdx.y) * 16 + tid) * KNN;
#pragma unroll
    for (int k = 0; k < KNN; ++k) {
      partD[pofs + k] = fd[k];
      partI[pofs + k] = fi[k];
    }
  }
}

// ---- stage 4: merge SPLIT partial lists per particle, radius mask ----------
__global__ __launch_bounds__(256) void tl_knn_merge(
    const float* __restrict__ partD, const int* __restrict__ partI,
    const float* __restrict__ avec,
    int* __restrict__ knn_idx, int M) {
  int m = blockIdx.x * blockDim.x + threadIdx.x;
  if (m >= M) return;
  float fd[KNN]; int fi[KNN];
#pragma unroll
  for (int k = 0; k < KNN; ++k) { fd[k] = 3.0e38f; fi[k] = 0; }
  for (int s = 0; s < SPLIT; ++s) {
    size_t pofs = (((size_t)(m >> 4) * SPLIT + s) * 16 + (m & 15)) * KNN;
    for (int k = 0; k < KNN; ++k) {
      float v = partD[pofs + k];
      if (v >= fd[KNN-1]) break;         // sorted ascending within a list
      int idxc = partI[pofs + k];
      fd[KNN-1] = v; fi[KNN-1] = idxc;
#pragma unroll
      for (int t = KNN-1; t > 0; --t) {
        if (fd[t] < fd[t-1]) {
          float fsw = fd[t]; fd[t] = fd[t-1]; fd[t-1] = fsw;
          int isw = fi[t]; fi[t] = fi[t-1]; fi[t-1] = isw;
        }
      }
    }
  }
  // radius mask needs true d2 = d2' + |p1|^2
  const float* ap = avec + (size_t)m * 4;
  float p1n = fmaf(ap[0], ap[0], fmaf(ap[1], ap[1], ap[2]*ap[2]));
#pragma unroll
  for (int k = 0; k < KNN; ++k) {
    knn_idx[(size_t)m * KNN + k] = (fd[k] + p1n <= RADIUS2) ? fi[k] : 0;
  }
}

// ---- stage 5: per-particle scoring ----------------------------------------
__global__ __launch_bounds__(256) void tl_score(
    const int*   __restrict__ knn,  const float* __restrict__ m2s,
    const float* __restrict__ dsrc, const float* __restrict__ co3,
    const float* __restrict__ m2t,  const float* __restrict__ dtgt,
    const float* __restrict__ sp,   const float* __restrict__ tp,
    float* __restrict__ per_particle, int M) {
  int m = blockIdx.x * blockDim.x + threadIdx.x;
  if (m >= M) return;
  int idx[KNN]; float dv[KNN];
#pragma unroll
  for (int k = 0; k < KNN; ++k) {
    idx[k] = knn[(size_t)m * KNN + k];
    dv[k]  = dsrc[idx[k]];
  }
  // stable insertion sort by source depth (matches jnp.argsort stability)
  for (int i = 1; i < KNN; ++i) {
    float kd = dv[i]; int ki = idx[i]; int j = i - 1;
    while (j >= 0 && dv[j] > kd) { dv[j+1] = dv[j]; idx[j+1] = idx[j]; --j; }
    dv[j+1] = kd; idx[j+1] = ki;
  }
  float spx = sp[2*m], spy = sp[2*m+1];
  float tpx = tp[2*m], tpy = tp[2*m+1];
  float run = 1.0f, total = 0.0f;
#pragma unroll
  for (int k = 0; k < KNN; ++k) {
    int g = idx[k];
    float dx = m2s[2*g]   - spx;
    float dy = m2s[2*g+1] - spy;
    float conA = co3[3*g], conC = co3[3*g+1], op = co3[3*g+2];
    // replicate reference exactly: -0.5*(conA + dx^2 + conC*dy^2), co[1]==0
    float power = -0.5f * (conA + dx*dx + conC*dy*dy);
    power = fminf(power, 0.0f);
    float alpha = fminf(fmaxf(op * __expf(power), 0.0f), 0.99f);
    if (k > 0) run *= (1.0f - alpha);    // agg/agg[:, :1] normalization
    float infl  = alpha * run;
    float d_src = sqrtf(dx*dx + dy*dy);
    float tdx = m2t[2*g]   - tpx;
    float tdy = m2t[2*g+1] - tpy;
    float d_tgt = sqrtf(tdx*tdx + tdy*tdy);
    total += fabsf(d_tgt * dtgt[g] - d_src * dv[k]) * infl;
  }
  per_particle[m] = total;
}

// ---- stage 6: mean ---------------------------------------------------------
__global__ __launch_bounds__(256) void tl_reduce(
    const float* __restrict__ per_particle, float* __restrict__ out, int M) {
  __shared__ float s[256];
  float acc = 0.0f;
  for (int i = threadIdx.x; i < M; i += 256) acc += per_particle[i];
  s[threadIdx.x] = acc;
  __syncthreads();
  for (int off = 128; off > 0; off >>= 1) {
    if (threadIdx.x < off) s[threadIdx.x] += s[threadIdx.x + off];
    __syncthreads();
  }
  if (threadIdx.x == 0) out[0] = s[0] / (float)M;
}

// ---------------------------------------------------------------------------
extern "C" void kernel_launch(void* const* d_in, const int* in_sizes, int n_in,
                              void* d_out, int out_size, void* d_ws, size_t ws_size,
                              hipStream_t stream) {
  (void)n_in; (void)out_size; (void)ws_size;
  const float* srcP = (const float*)d_in[0];
  const float* tgtP = (const float*)d_in[1];
  const float* srcX = (const float*)d_in[2];
  const float* tgtX = (const float*)d_in[3];
  const float* opac = (const float*)d_in[4];
  const float* scal = (const float*)d_in[5];
  const float* sp   = (const float*)d_in[6];
  const float* tp   = (const float*)d_in[7];
  const int*   seg  = (const int*)d_in[8];
  const int*   pseg = (const int*)d_in[9];
  const int N = in_sizes[2] / 3;   // 131072
  const int M = in_sizes[6] / 2;   // 2048
  const int MT = M / 16;           // particle tiles

  size_t off = 0;
  auto carve = [&](size_t nbytes) -> void* {
    void* p = (char*)d_ws + off;
    off += (nbytes + 255) & ~(size_t)255;
    return p;
  };
  float* bvec  = (float*)carve((size_t)N * 4 * sizeof(float));
  float* m2s   = (float*)carve((size_t)N * 2 * sizeof(float));
  float* dsrc  = (float*)carve((size_t)N * sizeof(float));
  float* co3   = (float*)carve((size_t)N * 3 * sizeof(float));
  float* m2t   = (float*)carve((size_t)N * 2 * sizeof(float));
  float* dtgt  = (float*)carve((size_t)N * sizeof(float));
  float* avec  = (float*)carve((size_t)M * 4 * sizeof(float));
  float* partD = (float*)carve((size_t)MT * SPLIT * 16 * KNN * sizeof(float));
  int*   partI = (int*)  carve((size_t)MT * SPLIT * 16 * KNN * sizeof(int));
  int*   knn   = (int*)  carve((size_t)M * KNN * sizeof(int));
  float* pp    = (float*)carve((size_t)M * sizeof(float));

  const int tpb = 256;
  tl_prep_n<<<(N + tpb - 1) / tpb, tpb, 0, stream>>>(
      srcP, tgtP, srcX, tgtX, opac, scal, seg,
      bvec, m2s, dsrc, co3, m2t, dtgt, N);
  tl_prep_m<<<(M + tpb - 1) / tpb, tpb, 0, stream>>>(sp, pseg, avec, M);
  dim3 kgrid(MT, SPLIT);
  tl_knn_part<<<kgrid, 256, 0, stream>>>(avec, bvec, partD, partI, N);
  tl_knn_merge<<<(M + tpb - 1) / tpb, tpb, 0, stream>>>(
      partD, partI, avec, knn, M);
  tl_score<<<(M + tpb - 1) / tpb, tpb, 0, stream>>>(
      knn, m2s, dsrc, co3, m2t, dtgt, sp, tp, pp, M);
  tl_reduce<<<1, 256, 0, stream>>>(pp, (float*)d_out, M);
}